// AddAttention_67482526155257
// MI455X (gfx1250) — compile-verified
//
#include <hip/hip_runtime.h>

#define Bn 8
#define Ln 1024
#define Hn 512
#define ROWP 516  // padded LDS row stride (floats) to avoid 64-bank conflicts

typedef __attribute__((ext_vector_type(2))) float v2f;
typedef __attribute__((ext_vector_type(8))) float v8f;
typedef __attribute__((ext_vector_type(4))) int   v4i;

typedef __attribute__((address_space(1))) v4i* as1_v4i_ptr;  // global
typedef __attribute__((address_space(3))) v4i* as3_v4i_ptr;  // LDS

#if defined(__gfx1250__) && __has_builtin(__builtin_amdgcn_global_load_async_to_lds_b128)
#define HAVE_ASYNC_LDS 1
#else
#define HAVE_ASYNC_LDS 0
#endif

__device__ __forceinline__ float tanh_fast(float x) {
#if __has_builtin(__builtin_amdgcn_tanhf)
  return __builtin_amdgcn_tanhf(x);
#else
  // tanh(x) = sign(x) * (1 - 2/(exp2(2|x|*log2e)+1)); saturates cleanly to +-1
  float ax = __builtin_fabsf(x);
  float e  = __builtin_amdgcn_exp2f(ax * 2.88539008177793f);  // 2*log2(e)
  float r  = 1.0f - 2.0f * __builtin_amdgcn_rcpf(e + 1.0f);
  return __builtin_copysignf(r, x);
#endif
}

// 16-byte global -> LDS copy; async DMA (ASYNCcnt) when the toolchain has it.
__device__ __forceinline__ void copy16_g2l(const float* __restrict__ g,
                                           float* __restrict__ l) {
#if HAVE_ASYNC_LDS
  __builtin_amdgcn_global_load_async_to_lds_b128(
      (as1_v4i_ptr)g, (as3_v4i_ptr)l, 0, 0);
#else
  *(float4*)l = *(const float4*)g;
#endif
}

__device__ __forceinline__ void wait_async_copies() {
#if HAVE_ASYNC_LDS
#if __has_builtin(__builtin_amdgcn_s_wait_asynccnt)
  __builtin_amdgcn_s_wait_asynccnt(0);
#else
  asm volatile("s_wait_asynccnt 0x0" ::: "memory");
#endif
#endif
}

// ---------------------------------------------------------------------------
// U[M,H] = X[M,H] @ W[H,H]^T + bias    (one wave computes one 16x16 tile)
// A frag (16x4 f32): lane = m + 16*(k>=2), vgpr v -> k = 2*(lane>>4) + v
// B frag (4x16 f32): lane = n + 16*(k>=2), vgpr v -> k = 2*(lane>>4) + v
//   B[k][n] = W[n][k]  -> both frags are contiguous-in-k b64 loads.
// ---------------------------------------------------------------------------
__global__ void gemm_xwT_bias(const float* __restrict__ X,
                              const float* __restrict__ W,
                              const float* __restrict__ bias,
                              float* __restrict__ U) {
  const int lane = threadIdx.x;     // 32 threads = 1 wave
  const int lm = lane & 15;
  const int lh = lane >> 4;         // 0 or 1
  const int m0 = blockIdx.x * 16;
  const int n0 = blockIdx.y * 16;

  const float* Ar = X + (size_t)(m0 + lm) * Hn + 2 * lh;
  const float* Br = W + (size_t)(n0 + lm) * Hn + 2 * lh;

  v8f acc = {};
#pragma unroll 4
  for (int k = 0; k < Hn; k += 4) {
    v2f a = *(const v2f*)(Ar + k);
    v2f b = *(const v2f*)(Br + k);
    acc = __builtin_amdgcn_wmma_f32_16x16x4_f32(false, a, false, b,
                                                (short)0, acc, false, false);
  }
  const float bv = bias[n0 + lm];
  // D layout: lane l, vgpr v -> row = v + 8*(l>>4), col = l&15
  float* out = U + (size_t)(m0 + 8 * lh) * Hn + n0 + lm;
#pragma unroll
  for (int v = 0; v < 8; ++v) out[(size_t)v * Hn] = acc[v] + bv;
}

// ---------------------------------------------------------------------------
// scores[b,i,j] = sum_h q[h] * tanh(u1[b,i,h] + u2[b,j,h])
// 256 threads: 32x16 (i,j) tile; each thread does (i,j) and (i+16,j),
// sharing the u2 row and q chunk. Tiles staged in LDS (async when available).
// ---------------------------------------------------------------------------
__global__ void scores_kernel(const float* __restrict__ U1,
                              const float* __restrict__ U2,
                              const float* __restrict__ q,
                              float* __restrict__ S) {
  __shared__ float s_u1[32 * ROWP];
  __shared__ float s_u2[16 * ROWP];
  __shared__ float s_q[Hn];

  const int t  = threadIdx.x;       // 0..255
  const int b  = blockIdx.z;
  const int i0 = blockIdx.y * 32;
  const int j0 = blockIdx.x * 16;

  const float* g1 = U1 + ((size_t)b * Ln + i0) * Hn;
  const float* g2 = U2 + ((size_t)b * Ln + j0) * Hn;

  // u1 tile: 32 rows x 128 float4 = 4096 -> 16 per thread
#pragma unroll
  for (int c = 0; c < 16; ++c) {
    int idx = t + c * 256;
    int r   = idx >> 7;             // row 0..31
    int c4  = idx & 127;            // float4 column
    copy16_g2l(g1 + (size_t)r * Hn + c4 * 4, s_u1 + r * ROWP + c4 * 4);
  }
  // u2 tile: 16 rows x 128 float4 = 2048 -> 8 per thread
#pragma unroll
  for (int c = 0; c < 8; ++c) {
    int idx = t + c * 256;
    int r   = idx >> 7;
    int c4  = idx & 127;
    copy16_g2l(g2 + (size_t)r * Hn + c4 * 4, s_u2 + r * ROWP + c4 * 4);
  }
  if (t < 128) copy16_g2l(q + t * 4, s_q + t * 4);
  wait_async_copies();
  __syncthreads();

  const int j = t & 15;
  const int i = t >> 4;             // 0..15; second row is i+16
  const float4* r1a = (const float4*)(s_u1 + i * ROWP);
  const float4* r1b = (const float4*)(s_u1 + (i + 16) * ROWP);
  const float4* r2  = (const float4*)(s_u2 + j * ROWP);
  const float4* rq  = (const float4*)s_q;

  float acc0 = 0.0f, acc1 = 0.0f;
#pragma unroll 4
  for (int h4 = 0; h4 < Hn / 4; ++h4) {
    float4 xa = r1a[h4];
    float4 xb = r1b[h4];
    float4 x2 = r2[h4];
    float4 qq = rq[h4];
    acc0 = fmaf(qq.x, tanh_fast(xa.x + x2.x), acc0);
    acc1 = fmaf(qq.x, tanh_fast(xb.x + x2.x), acc1);
    acc0 = fmaf(qq.y, tanh_fast(xa.y + x2.y), acc0);
    acc1 = fmaf(qq.y, tanh_fast(xb.y + x2.y), acc1);
    acc0 = fmaf(qq.z, tanh_fast(xa.z + x2.z), acc0);
    acc1 = fmaf(qq.z, tanh_fast(xb.z + x2.z), acc1);
    acc0 = fmaf(qq.w, tanh_fast(xa.w + x2.w), acc0);
    acc1 = fmaf(qq.w, tanh_fast(xb.w + x2.w), acc1);
  }
  S[((size_t)b * Ln + (i0 + i)) * Ln + j0 + j] = acc0;
  S[((size_t)b * Ln + (i0 + i + 16)) * Ln + j0 + j] = acc1;
}

// ---------------------------------------------------------------------------
// In-place softmax over rows of length 1024 (256 threads, float4 per thread).
// ---------------------------------------------------------------------------
__global__ void softmax_rows(float* __restrict__ A) {
  const int t = threadIdx.x;
  float* p = A + (size_t)blockIdx.x * Ln;
  float4 v = ((float4*)p)[t];

  __shared__ float red[256];
  red[t] = fmaxf(fmaxf(v.x, v.y), fmaxf(v.z, v.w));
  __syncthreads();
  for (int s = 128; s > 0; s >>= 1) {
    if (t < s) red[t] = fmaxf(red[t], red[t + s]);
    __syncthreads();
  }
  const float rowmax = red[0];
  __syncthreads();

  const float c = 1.44269504088896f;  // log2(e)
  float4 e;
  e.x = __builtin_amdgcn_exp2f((v.x - rowmax) * c);
  e.y = __builtin_amdgcn_exp2f((v.y - rowmax) * c);
  e.z = __builtin_amdgcn_exp2f((v.z - rowmax) * c);
  e.w = __builtin_amdgcn_exp2f((v.w - rowmax) * c);

  red[t] = e.x + e.y + e.z + e.w;
  __syncthreads();
  for (int s = 128; s > 0; s >>= 1) {
    if (t < s) red[t] += red[t + s];
    __syncthreads();
  }
  const float inv = 1.0f / red[0];
  e.x *= inv; e.y *= inv; e.z *= inv; e.w *= inv;
  ((float4*)p)[t] = e;
}

// ---------------------------------------------------------------------------
// h[b] = a[b] @ inputs[b]   (M=L, K=L, N=H), one wave per 16x16 tile.
// ---------------------------------------------------------------------------
__global__ void gemm_av(const float* __restrict__ Amat,
                        const float* __restrict__ V,
                        float* __restrict__ Hout) {
  const int lane = threadIdx.x;
  const int lm = lane & 15;
  const int lh = lane >> 4;
  const int m0 = blockIdx.x * 16;
  const int n0 = blockIdx.y * 16;
  const int b  = blockIdx.z;

  const float* Ar = Amat + ((size_t)b * Ln + m0 + lm) * Ln + 2 * lh;
  const float* Br = V + (size_t)b * Ln * Hn + (size_t)(2 * lh) * Hn + n0 + lm;

  v8f acc = {};
#pragma unroll 4
  for (int k = 0; k < Ln; k += 4) {
    v2f a = *(const v2f*)(Ar + k);
    v2f bf;
    bf.x = Br[(size_t)k * Hn];        // row k + 2*lh
    bf.y = Br[(size_t)k * Hn + Hn];   // row k + 2*lh + 1
    acc = __builtin_amdgcn_wmma_f32_16x16x4_f32(false, a, false, bf,
                                                (short)0, acc, false, false);
  }
  float* out = Hout + ((size_t)b * Ln + m0 + 8 * lh) * Hn + n0 + lm;
#pragma unroll
  for (int v = 0; v < 8; ++v) out[(size_t)v * Hn] = acc[v];
}

// ---------------------------------------------------------------------------
extern "C" void kernel_launch(void* const* d_in, const int* in_sizes, int n_in,
                              void* d_out, int out_size, void* d_ws, size_t ws_size,
                              hipStream_t stream) {
  const float* inp = (const float*)d_in[0];  // [B,L,H]
  const float* W1  = (const float*)d_in[1];  // [H,H]
  const float* b1  = (const float*)d_in[2];  // [H]
  const float* W2  = (const float*)d_in[3];  // [H,H]
  const float* b2  = (const float*)d_in[4];  // [H]
  const float* q   = (const float*)d_in[5];  // [H]

  float* h_out = (float*)d_out;                       // [B,L,H]
  float* a_out = h_out + (size_t)Bn * Ln * Hn;        // [B,L,L]

  float* u1 = (float*)d_ws;                           // 16 MB
  float* u2 = u1 + (size_t)Bn * Ln * Hn;              // 16 MB  (needs ws >= 32MB)

  // u1 = X@W1^T + b1 ; u2 = X@W2^T + b2   (X flattened to [B*L, H])
  dim3 gU(Bn * Ln / 16, Hn / 16);                     // 512 x 32 tiles
  hipLaunchKernelGGL(gemm_xwT_bias, gU, dim3(32), 0, stream, inp, W1, b1, u1);
  hipLaunchKernelGGL(gemm_xwT_bias, gU, dim3(32), 0, stream, inp, W2, b2, u2);

  // raw scores straight into d_out's `a` region
  hipLaunchKernelGGL(scores_kernel, dim3(Ln / 16, Ln / 32, Bn), dim3(256), 0,
                     stream, u1, u2, q, a_out);

  // softmax in place over j
  hipLaunchKernelGGL(softmax_rows, dim3(Bn * Ln), dim3(256), 0, stream, a_out);

  // h = a @ inputs
  hipLaunchKernelGGL(gemm_av, dim3(Ln / 16, Hn / 16, Bn), dim3(32), 0, stream,
                     a_out, inp, h_out);
}